// Splatting_8203387535928
// MI455X (gfx1250) — compile-verified
//
#include <hip/hip_runtime.h>

// Problem shape (from reference setup_inputs)
namespace {
constexpr int B = 4, C = 16, H = 512, W = 1024;
constexpr int N = H * W;

// Tiling: 128x64 source pixels per workgroup, 16-px halo (3.2 sigma of the
// 5-px-std flow -> >99.7% of splats land inside the LDS tile).
constexpr int TW = 128, TH = 64;
constexpr int TILE = TW * TH;         // 8192 source pixels
constexpr int R  = 16;
constexpr int RW = TW + 2 * R;        // 160 (float4-aligned rows)
constexpr int RH = TH + 2 * R;        // 96
constexpr int REGION = RW * RH;       // 15360 cells -> 60 KB accumulator
constexpr int THREADS = 256;          // 8 wave32 waves
constexpr int PIX_PER_THREAD = TILE / THREADS;    // 32
constexpr int NV4 = REGION / 4;       // 3840 = 15 * THREADS (exact)
}

// Device-scope, no-return FP32 atomic add straight to the ISA instruction.
// Tracked by STOREcnt; drained by the implicit wait-idle at S_ENDPGM (we
// never read `out` in this kernel).
__device__ __forceinline__ void global_atomic_fadd(float* p, float v) {
  asm volatile("global_atomic_add_f32 %0, %1, off scope:SCOPE_DEV"
               :: "v"(p), "v"(v)
               : "memory");
}

// CDNA5 async copy: global -> LDS, 16 bytes per lane, tracked by ASYNCcnt.
// The LDS destination operand is the workgroup-relative LDS byte address,
// i.e. the low 32 bits of the flat shared-aperture address (ISA 10.2).
__device__ __forceinline__ void async_load_lds_b128(uint32_t lds_byte_off,
                                                    const float* gptr) {
  asm volatile("global_load_async_to_lds_b128 %0, %1, off"
               :: "v"(lds_byte_off), "v"(gptr)
               : "memory");
}

__device__ __forceinline__ void wait_asynccnt0() {
  asm volatile("s_wait_asynccnt 0x0" ::: "memory");
}

__global__ __launch_bounds__(THREADS)
void splat_tiled_kernel(const float* __restrict__ frame,
                        const float* __restrict__ flow,
                        float* __restrict__ out) {
  // 60 KB accumulator + 64 KB staged flow = 124 KB of the 320 KB WGP LDS
  // -> 2 resident workgroups per WGP.
  __shared__ __align__(16) float acc[REGION];
  __shared__ __align__(16) float sflow[2 * TILE];  // [0,TILE)=dx, [TILE,2T)=dy

  const int tileX = blockIdx.x * TW;
  const int tileY = blockIdx.y * TH;
  const int b     = blockIdx.z;
  const int t     = threadIdx.x;

  const float* __restrict__ flowx = flow + (size_t)b * 2 * N;   // flow[b,0]
  const float* __restrict__ flowy = flowx + N;                  // flow[b,1]

  // ---- Stage the tile's flow into LDS once (async DMA, B128/lane) -------
  for (int j4 = t; j4 < TILE / 4; j4 += THREADS) {
    const int j  = j4 * 4;
    const int py = j / TW;           // TW=128
    const int px = j % TW;           // multiple of 4 -> rows stay aligned
    const size_t sp = (size_t)(tileY + py) * W + (tileX + px);
    async_load_lds_b128((uint32_t)(uintptr_t)&sflow[j],        flowx + sp);
    async_load_lds_b128((uint32_t)(uintptr_t)&sflow[TILE + j], flowy + sp);
  }
  wait_asynccnt0();   // per-wave drain; cross-wave visibility via barrier below

  // ---- One channel per pass: 60 KB accumulator, 16 passes ---------------
  for (int c = 0; c < C; ++c) {
    const float* __restrict__ fch = frame + (size_t)(b * C + c) * N;
    float* __restrict__ och       = out   + (size_t)(b * C + c) * N;

    // 1) zero the LDS accumulator tile (b128 stores, perfectly balanced)
    for (int jv = t; jv < NV4; jv += THREADS) {
      reinterpret_cast<float4*>(acc)[jv] = make_float4(0.f, 0.f, 0.f, 0.f);
    }
    __syncthreads();   // also publishes sflow on the first pass

    // 2) splat this tile's source pixels into LDS
    for (int i = 0; i < PIX_PER_THREAD; ++i) {
      const int pid = t + i * THREADS;
      const int px  = pid % TW;            // consecutive threads -> coalesced
      const int py  = pid / TW;
      const int gx  = tileX + px;
      const int gy  = tileY + py;
      const size_t sp = (size_t)gy * W + gx;

      const float dxf = sflow[pid];        // staged flow, LDS-resident
      const float dyf = sflow[TILE + pid];
      const float f   = fch[sp];

      // Warm L2 for the next channel pass (global_prefetch_b8).
      if (c + 1 < C) __builtin_prefetch(fch + N + sp, 0, 0);

      const float tx  = (float)gx + dxf;
      const float ty  = (float)gy + dyf;
      const float x0f = floorf(tx);
      const float y0f = floorf(ty);
      const float fx  = tx - x0f;
      const float fy  = ty - y0f;
      const int   x0  = (int)x0f;
      const int   y0  = (int)y0f;

      const float wx[2] = {1.0f - fx, fx};
      const float wy[2] = {1.0f - fy, fy};

#pragma unroll
      for (int k = 0; k < 4; ++k) {
        const int cdy = k >> 1;
        const int cdx = k & 1;
        const int xi = x0 + cdx;
        const int yi = y0 + cdy;
        // Reference zeroes the weight for out-of-image corners -> skip.
        if (xi < 0 || xi >= W || yi < 0 || yi >= H) continue;
        const float contrib = f * (wx[cdx] * wy[cdy]);
        const int lx = xi - (tileX - R);
        const int ly = yi - (tileY - R);
        if ((unsigned)lx < (unsigned)RW && (unsigned)ly < (unsigned)RH) {
          // Hot path: LDS float atomic (ds_add_f32), 64 banks per WGP.
          atomicAdd(&acc[ly * RW + lx], contrib);
        } else {
          // Rare Gaussian-tail path: direct device-scope atomic.
          global_atomic_fadd(och + (size_t)yi * W + xi, contrib);
        }
      }
    }
    __syncthreads();   // drains DScnt before we read acc[]

    // 3) flush tile: one device-scope atomic per live cell (b128 LDS reads)
    for (int jv = t; jv < NV4; jv += THREADS) {
      const float4 v4 = reinterpret_cast<const float4*>(acc)[jv];
      const int cell0 = jv * 4;
      const int ly    = cell0 / RW;        // RW%4==0 -> whole vector one row
      const int lx0   = cell0 - ly * RW;
      const int gyy   = tileY - R + ly;
      if ((unsigned)gyy < (unsigned)H) {
        float* rowp = och + (size_t)gyy * W + (tileX - R + lx0);
        const float vv[4] = {v4.x, v4.y, v4.z, v4.w};
#pragma unroll
        for (int e = 0; e < 4; ++e) {
          const int gxx = tileX - R + lx0 + e;
          if (vv[e] != 0.0f && (unsigned)gxx < (unsigned)W) {
            global_atomic_fadd(rowp + e, vv[e]);
          }
        }
      }
    }
    __syncthreads();   // before zeroing for the next channel pass
  }
}

extern "C" void kernel_launch(void* const* d_in, const int* in_sizes, int n_in,
                              void* d_out, int out_size, void* d_ws, size_t ws_size,
                              hipStream_t stream) {
  const float* frame = (const float*)d_in[0];  // [B,C,H,W] f32
  const float* flow  = (const float*)d_in[1];  // [B,2,H,W] f32
  float* out = (float*)d_out;                  // [B,C,H,W] f32

  // Scatter-add target must start from zero (harness poisons d_out).
  hipMemsetAsync(out, 0, (size_t)out_size * sizeof(float), stream);

  dim3 grid(W / TW, H / TH, B);   // 8 x 8 x 4 = 256 workgroups
  splat_tiled_kernel<<<grid, dim3(THREADS), 0, stream>>>(frame, flow, out);
}